// TankLogicValueNetwork_89292370084186
// MI455X (gfx1250) — compile-verified
//
#include <hip/hip_runtime.h>
#include <hip/hip_bf16.h>

#define BROWS   4096
#define IN_DIM  1024
#define HIDDEN  8192
#define N_LAYERS 6
#define OUT_DIM 8
#define NPO     (HIDDEN / OUT_DIM)

#define ROWS    4      // batch rows per workgroup (LDS-resident)
#define THREADS 512

typedef __attribute__((ext_vector_type(2))) float v2f;
typedef __attribute__((ext_vector_type(8))) float v8f;

__constant__ float GC[16][4] = {
    {0.f, 0.f, 0.f, 0.f}, {0.f, 0.f, 0.f, 1.f}, {0.f, 1.f, 0.f, -1.f}, {0.f, 1.f, 0.f, 0.f},
    {0.f, 0.f, 1.f, -1.f}, {0.f, 0.f, 1.f, 0.f}, {0.f, 1.f, 1.f, -2.f}, {0.f, 1.f, 1.f, -1.f},
    {1.f, -1.f, -1.f, 1.f}, {1.f, -1.f, -1.f, 2.f}, {1.f, 0.f, -1.f, 0.f}, {1.f, 0.f, -1.f, 1.f},
    {1.f, -1.f, 0.f, 0.f}, {1.f, -1.f, 0.f, 1.f}, {1.f, 0.f, 0.f, -1.f}, {1.f, 0.f, 0.f, 0.f}};

// Kernel 1: fold softmax(w) @ GATE_COEF into 4 coefficients per neuron.
__global__ __launch_bounds__(256) void coef_kernel(const float* __restrict__ logic_w,
                                                   float* __restrict__ coef) {
    int n = blockIdx.x * blockDim.x + threadIdx.x;
    if (n >= N_LAYERS * HIDDEN) return;
    const float* w = logic_w + (size_t)n * 16;
    float wv[16], m = -INFINITY;
#pragma unroll
    for (int g = 0; g < 16; ++g) { wv[g] = w[g]; m = fmaxf(m, wv[g]); }
    float p[16], s = 0.f;
#pragma unroll
    for (int g = 0; g < 16; ++g) { p[g] = __expf(wv[g] - m); s += p[g]; }
    float inv = 1.f / s;
    float c0 = 0.f, c1 = 0.f, c2 = 0.f, c3 = 0.f;
#pragma unroll
    for (int g = 0; g < 16; ++g) {
        float pg = p[g] * inv;
        c0 += pg * GC[g][0]; c1 += pg * GC[g][1];
        c2 += pg * GC[g][2]; c3 += pg * GC[g][3];
    }
    float4 o; o.x = c0; o.y = c1; o.z = c2; o.w = c3;
    ((float4*)coef)[n] = o;
}

// Kernel 2: LDS-resident logic layers + per-row partial-sum reduction.
__global__ __launch_bounds__(THREADS) void logic_kernel(
    const float* __restrict__ x,
    const int* __restrict__ idx0_a, const int* __restrict__ idx0_b,
    const int* __restrict__ idx_a, const int* __restrict__ idx_b,
    const float4* __restrict__ coef, float* __restrict__ values) {
    extern __shared__ float smem[];
    float* bin  = smem;                    // ROWS * IN_DIM
    float* bufA = bin + ROWS * IN_DIM;     // ROWS * HIDDEN
    float* bufB = bufA + ROWS * HIDDEN;    // ROWS * HIDDEN

    const int tid  = threadIdx.x;
    const int row0 = blockIdx.x * ROWS;

    // Binarize x into LDS.
    for (int i = tid; i < ROWS * IN_DIM; i += THREADS)
        bin[i] = (x[(size_t)row0 * IN_DIM + i] > 0.5f) ? 1.0f : 0.0f;
    __syncthreads();

    // Layer 0: gather from binary inputs.
    for (int j = tid; j < HIDDEN; j += THREADS) {
        int ia = idx0_a[j], ib = idx0_b[j];
        float4 c = coef[j];
#pragma unroll
        for (int r = 0; r < ROWS; ++r) {
            float a = bin[r * IN_DIM + ia];
            float b = bin[r * IN_DIM + ib];
            bufA[r * HIDDEN + j] = c.x + c.y * a + c.z * b + c.w * (a * b);
        }
    }
    __syncthreads();

    // Layers 1..5: ping-pong LDS buffers.
    float* src = bufA;
    float* dst = bufB;
    for (int l = 1; l < N_LAYERS; ++l) {
        const int* pa = idx_a + (size_t)(l - 1) * HIDDEN;
        const int* pb = idx_b + (size_t)(l - 1) * HIDDEN;
        const float4* pc = coef + (size_t)l * HIDDEN;
        for (int j = tid; j < HIDDEN; j += THREADS) {
            int ia = pa[j], ib = pb[j];
            float4 c = pc[j];
#pragma unroll
            for (int r = 0; r < ROWS; ++r) {
                float a = src[r * HIDDEN + ia];
                float b = src[r * HIDDEN + ib];
                dst[r * HIDDEN + j] = c.x + c.y * a + c.z * b + c.w * (a * b);
            }
        }
        __syncthreads();
        float* t = src; src = dst; dst = t;   // final result ends in `src`
    }

    // Reduce 8192 -> 8 per row: 16-lane groups per (row, out) pair.
    int r    = tid >> 7;        // 0..3
    int o    = (tid >> 4) & 7;  // 0..7
    int lane = tid & 15;
    const float* hrow = src + r * HIDDEN + o * NPO;
    float s = 0.f;
    for (int k = lane; k < NPO; k += 16) s += hrow[k];
#pragma unroll
    for (int off = 8; off >= 1; off >>= 1) s += __shfl_xor(s, off, 16);
    if (lane == 0) values[(size_t)(row0 + r) * OUT_DIM + o] = s * 0.1f;
}

// Kernel 3: q = values @ scaler_w^T + scaler_b via V_WMMA_F32_16X16X4_F32.
// One wave handles 16 batch rows; K=8 accumulated as two K=4 WMMAs.
__global__ __launch_bounds__(32) void qhead_kernel(const float* __restrict__ values,
                                                   const float* __restrict__ scaler_w,
                                                   const float* __restrict__ scaler_b,
                                                   float* __restrict__ out) {
    const int lane = threadIdx.x;       // 0..31
    const int half = lane >> 4;         // lane group
    const int m    = lane & 15;
    const int row0 = blockIdx.x * 16;

    // A fragment (16x4, f32): lanes 0-15 hold K=0(v0),K=1(v1); lanes 16-31 K=2,K=3.
    v2f a0, a1;
    const float* vr = values + (size_t)(row0 + m) * OUT_DIM;
    a0.x = vr[2 * half + 0];
    a0.y = vr[2 * half + 1];
    a1.x = vr[4 + 2 * half + 0];
    a1.y = vr[4 + 2 * half + 1];

    // B fragment (4x16): B[k][n] = scaler_w[n][k]; rows striped across lanes.
    v2f b0, b1;
    int n = m;
    if (n < OUT_DIM) {
        const float* wr = scaler_w + (size_t)n * OUT_DIM;
        b0.x = wr[2 * half + 0];
        b0.y = wr[2 * half + 1];
        b1.x = wr[4 + 2 * half + 0];
        b1.y = wr[4 + 2 * half + 1];
    } else {
        b0.x = 0.f; b0.y = 0.f; b1.x = 0.f; b1.y = 0.f;
    }

    v8f c = {};
    c = __builtin_amdgcn_wmma_f32_16x16x4_f32(false, a0, false, b0, (short)0, c, false, false);
    c = __builtin_amdgcn_wmma_f32_16x16x4_f32(false, a1, false, b1, (short)0, c, false, false);

    // D layout: VGPR v, lane l -> row = v + 8*(l>>4), col = l&15.
    if (n < OUT_DIM) {
        float bias = scaler_b[n];
#pragma unroll
        for (int v = 0; v < 8; ++v) {
            int row = row0 + v + 8 * half;
            out[(size_t)row * OUT_DIM + n] = c[v] + bias;
        }
    }
}

extern "C" void kernel_launch(void* const* d_in, const int* in_sizes, int n_in,
                              void* d_out, int out_size, void* d_ws, size_t ws_size,
                              hipStream_t stream) {
    const float* x        = (const float*)d_in[0];
    const float* logic_w  = (const float*)d_in[1];
    const float* scaler_w = (const float*)d_in[2];
    const float* scaler_b = (const float*)d_in[3];
    const int*   idx0_a   = (const int*)d_in[4];
    const int*   idx0_b   = (const int*)d_in[5];
    const int*   idx_a    = (const int*)d_in[6];
    const int*   idx_b    = (const int*)d_in[7];

    float* coef   = (float*)d_ws;                          // 6*8192*4 floats
    float* values = coef + (size_t)N_LAYERS * HIDDEN * 4;  // 4096*8 floats
    float* out    = (float*)d_out;

    coef_kernel<<<(N_LAYERS * HIDDEN + 255) / 256, 256, 0, stream>>>(logic_w, coef);

    size_t shmem = (size_t)ROWS * (IN_DIM + 2 * HIDDEN) * sizeof(float);  // ~272 KB (<320 KB/WGP)
    (void)hipFuncSetAttribute(reinterpret_cast<const void*>(logic_kernel),
                              hipFuncAttributeMaxDynamicSharedMemorySize, (int)shmem);
    logic_kernel<<<BROWS / ROWS, THREADS, shmem, stream>>>(x, idx0_a, idx0_b, idx_a, idx_b,
                                                           (const float4*)coef, values);

    qhead_kernel<<<BROWS / 16, 32, 0, stream>>>(values, scaler_w, scaler_b, out);
}